// GCN_29523605193127
// MI455X (gfx1250) — compile-verified
//
#include <hip/hip_runtime.h>

#define NN 100000      // nodes
#define NE 1600000     // edges
#define FEATS 128      // in feats == hidden
#define LDS_STRIDE 144 // padded row stride in floats: 2*144 % 64banks = 32 -> conflict-free halves

typedef __attribute__((ext_vector_type(2))) float v2f;
typedef __attribute__((ext_vector_type(8))) float v8f;
typedef __attribute__((address_space(3))) float lds_f;
typedef __attribute__((address_space(3))) char  lds_c;

// ---------------- zero workspace (agg + both degree arrays) ----------------
__global__ void gcn_zero(int* __restrict__ ws, int nWords) {
  int i = blockIdx.x * blockDim.x + threadIdx.x;
  int stride = gridDim.x * blockDim.x;
  for (; i < nWords; i += stride) ws[i] = 0;
}

// ---------------- degree counting: 2 int atomics per edge ------------------
__global__ void gcn_count_deg(const int* __restrict__ src, const int* __restrict__ dst,
                              int* __restrict__ deg_out, int* __restrict__ deg_in, int nE) {
  int e = blockIdx.x * blockDim.x + threadIdx.x;
  if (e < nE) {
    atomicAdd(&deg_out[src[e]], 1);
    atomicAdd(&deg_in[dst[e]], 1);
  }
}

// ---------------- deg -> rsqrt(max(deg,1)) in place ------------------------
__global__ void gcn_deg_rsqrt(int* __restrict__ deg_out, int* __restrict__ deg_in, int n) {
  int i = blockIdx.x * blockDim.x + threadIdx.x;
  if (i < n) {
    int a = deg_out[i];
    int b = deg_in[i];
    ((float*)deg_out)[i] = rsqrtf((float)(a < 1 ? 1 : a));
    ((float*)deg_in)[i]  = rsqrtf((float)(b < 1 ? 1 : b));
  }
}

// ---------------- scatter: one wave32 per edge -----------------------------
__global__ void __launch_bounds__(256) gcn_scatter(
    const float4* __restrict__ feat4, const int* __restrict__ src,
    const int* __restrict__ dst, const float* __restrict__ rs_out,
    float* __restrict__ agg, int nE) {
  int e = blockIdx.x * 8 + (threadIdx.x >> 5);
  if (e >= nE) return;
  int lane = threadIdx.x & 31;
  int s = src[e];
  int d = dst[e];
  float sc = rs_out[s];                       // fold src-side norm into gather
  float4 f = feat4[(size_t)s * 32 + lane];
  float* ap = agg + (size_t)d * FEATS + lane * 4;
  atomicAdd(ap + 0, f.x * sc);
  atomicAdd(ap + 1, f.y * sc);
  atomicAdd(ap + 2, f.z * sc);
  atomicAdd(ap + 3, f.w * sc);
}

// ---------------- fp32 WMMA GEMM + dst-norm + bias + ReLU ------------------
// out[m, n] = relu( rs_in[m] * agg[m, :] @ W[:, n] + bias[n] )
// W (64KB) staged to LDS with CDNA5 async global->LDS copies (ASYNCcnt path).
// LDS destination passed as a real addrspace(3) pointer derived from sW so
// the compiler keeps sW alive and treats the asm as a writer of it.
__global__ void __launch_bounds__(256) gcn_gemm_bias_relu(
    const float* __restrict__ agg, const float* __restrict__ rs_in,
    const float* __restrict__ W, const float* __restrict__ bias,
    float* __restrict__ out) {
  __shared__ float sW[FEATS * LDS_STRIDE];  // 72 KB, padded rows

  const int lane = threadIdx.x & 31;
  const int wave = threadIdx.x >> 5;

  // --- async-stage W: wave w copies rows r = w + 8*i, one 512B row / inst --
  {
    lds_f* sW3 = (lds_f*)sW;                // addrspacecast generic -> LDS(3)
    lds_c* lp0 = (lds_c*)sW3 + (wave * (LDS_STRIDE * 4) + lane * 16);
    unsigned long long ga =
        (unsigned long long)W + (unsigned)(wave * 512 + lane * 16);
#pragma unroll
    for (int i = 0; i < 16; ++i) {
      lds_c* lp = lp0 + i * (8 * LDS_STRIDE * 4);                 // +4608B/iter
      unsigned long long go = ga + (unsigned long long)(i * 8 * 512); // +4096B
      asm volatile("global_load_async_to_lds_b128 %0, %1, off"
                   :: "v"(lp), "v"(go)
                   : "memory");
    }
    asm volatile("s_wait_asynccnt 0" ::: "memory");
  }
  __syncthreads();

  const int l    = lane & 15;
  const int half = lane >> 4;          // 0 or 1
  const int rowBase = blockIdx.x * 16; // 6250 row tiles exactly cover 100000
  const int colBase = wave * 16;

  const int arow = rowBase + l;
  const float scale = rs_in[arow];     // fold dst-side norm into A fragment
  const float* __restrict__ A = agg + (size_t)arow * FEATS;
  const float* B = sW + colBase + l;

  v8f c = {};
#pragma unroll 4
  for (int k = 0; k < FEATS; k += 4) {
    const int kk = k + half * 2;
    v2f a;
    a.x = A[kk] * scale;
    a.y = A[kk + 1] * scale;
    v2f b;
    b.x = B[kk * LDS_STRIDE];
    b.y = B[(kk + 1) * LDS_STRIDE];
    c = __builtin_amdgcn_wmma_f32_16x16x4_f32(
        /*neg_a=*/false, a, /*neg_b=*/false, b,
        /*c_mod=*/(short)0, c, /*reuse_a=*/false, /*reuse_b=*/false);
  }

  const int outCol = colBase + l;
  const float bv = bias[outCol];
#pragma unroll
  for (int j = 0; j < 8; ++j) {
    const int m = rowBase + j + half * 8;
    float v = c[j] + bv;
    out[(size_t)m * FEATS + outCol] = v > 0.f ? v : 0.f;
  }
}

extern "C" void kernel_launch(void* const* d_in, const int* in_sizes, int n_in,
                              void* d_out, int out_size, void* d_ws, size_t ws_size,
                              hipStream_t stream) {
  const float* features = (const float*)d_in[0];
  const int*   src      = (const int*)d_in[1];
  const int*   dst      = (const int*)d_in[2];
  const float* weight   = (const float*)d_in[3];
  const float* bias     = (const float*)d_in[4];
  float*       out      = (float*)d_out;

  // workspace layout: [agg: NN*128 f32][deg_out/rs_out: NN][deg_in/rs_in: NN]
  float* agg     = (float*)d_ws;
  int*   deg_out = (int*)(agg + (size_t)NN * FEATS);
  int*   deg_in  = deg_out + NN;

  const int nWords = NN * FEATS + 2 * NN;
  gcn_zero<<<4096, 256, 0, stream>>>((int*)d_ws, nWords);
  gcn_count_deg<<<(NE + 255) / 256, 256, 0, stream>>>(src, dst, deg_out, deg_in, NE);
  gcn_deg_rsqrt<<<(NN + 255) / 256, 256, 0, stream>>>(deg_out, deg_in, NN);
  gcn_scatter<<<(NE + 7) / 8, 256, 0, stream>>>(
      (const float4*)features, src, dst, (const float*)deg_out, agg, NE);
  gcn_gemm_bias_relu<<<NN / 16, 256, 0, stream>>>(
      agg, (const float*)deg_in, weight, bias, out);
}